// SymmetryFunctions_1932735284042
// MI455X (gfx1250) — compile-verified
//
#include <hip/hip_runtime.h>

typedef __attribute__((ext_vector_type(2))) float v2f;
typedef __attribute__((ext_vector_type(8))) float v8f;

#define N_RAD    32
#define N_MU_ANG 12
#define N_BASIS  56
#define ETA_RAD  4.0f
#define ETA_ANG  8.0f
#define CUTOFF   5.0f
#define PI_F     3.14159265358979323846f

__device__ __forceinline__ float fcut(float d) {
    float fc = 0.5f * (__cosf(PI_F * (1.0f / CUTOFF) * d) + 1.0f);
    return (d < CUTOFF) ? fc : 0.0f;
}

// ---------------------------------------------------------------------------
// Pass 0: zero the accumulator (d_out itself, fp32)
// ---------------------------------------------------------------------------
__global__ void __launch_bounds__(256) sf_zero(float* __restrict__ out, int n) {
    int i = blockIdx.x * blockDim.x + threadIdx.x;
    if (i < n) out[i] = 0.0f;
}

// ---------------------------------------------------------------------------
// Pass 1: pairs. Each wave handles 16 pairs; lanes L and L+16 mirror pair L%16.
// Writes per-pair invariants {ux,uy,uz,d} and w = Z_j * fc(d) to workspace,
// and accumulates the 32 radial features via two 16x16x4 f32 WMMAs that build
// the Gaussian exponent matrix  E[p,m] = -eta*(d_p - mu_m)^2.
// ---------------------------------------------------------------------------
__global__ void __launch_bounds__(256) sf_pairs(
    const int* __restrict__ Z, const float* __restrict__ r_ij,
    const int* __restrict__ idx_i, const int* __restrict__ idx_j,
    const float* __restrict__ mus_rad,
    float4* __restrict__ pd, float* __restrict__ wArr,
    float* __restrict__ out, int n_pairs)
{
    const int tid  = blockIdx.x * blockDim.x + threadIdx.x;
    const int lane = tid & 31;
    const int wave = tid >> 5;
    const int m    = lane & 15;          // column (mu) / pair slot
    const bool lo  = lane < 16;

    int  p  = wave * 16 + m;
    bool v  = p < n_pairs;
    int  pc = v ? p : 0;

    float rx = r_ij[3 * pc + 0];
    float ry = r_ij[3 * pc + 1];
    float rz = r_ij[3 * pc + 2];
    float d2 = rx * rx + ry * ry + rz * rz;
    float d  = sqrtf(d2);
    float id = __frcp_rn(d);
    float w  = v ? fcut(d) * (float)Z[idx_j[pc]] : 0.0f;
    int   ai = idx_i[pc];

    if (v && lo) {
        pd[p]   = make_float4(rx * id, ry * id, rz * id, d);
        wArr[p] = w;
    }

    // A (16x4, f32): lanes 0-15 -> (K0,K1) = (-eta*d^2, d); lanes 16-31 -> (K2,K3)=(1,0)
    v2f a;
    a.x = lo ? (-ETA_RAD) * d * d : 1.0f;
    a.y = lo ? d : 0.0f;

    #pragma unroll
    for (int tile = 0; tile < 2; ++tile) {
        float mu = mus_rad[tile * 16 + m];
        // B (4x16): col m = [1, 2*eta*mu, -eta*mu^2, 0]
        v2f b;
        b.x = lo ? 1.0f : (-ETA_RAD) * mu * mu;
        b.y = lo ? (2.0f * ETA_RAD) * mu : 0.0f;
        v8f c = {};
        c = __builtin_amdgcn_wmma_f32_16x16x4_f32(
                false, a, false, b, (short)0, c, false, false);
        const int col = tile * 16 + m;
        #pragma unroll
        for (int r = 0; r < 8; ++r) {
            int   prow = r + (lo ? 0 : 8);          // D row = pair slot
            float wp   = __shfl(w,  prow, 32);
            int   aip  = __shfl(ai, prow, 32);
            float val  = wp * __expf(c[r]);
            atomicAdd(&out[aip * N_BASIS + col], val);
        }
    }
}

// ---------------------------------------------------------------------------
// Pass 2: triples. Each wave handles 16 triples. Gathers the L2-resident
// per-pair records, builds the angular exponent tile with one WMMA,
// (1 +/- cos)^8 via 3 squarings, scatters 24 features per triple.
// ---------------------------------------------------------------------------
__global__ void __launch_bounds__(256) sf_triples(
    const int* __restrict__ idx_i_t, const int* __restrict__ idx_j_t,
    const int* __restrict__ idx_k_t, const float* __restrict__ mus_ang,
    const float4* __restrict__ pd, const float* __restrict__ wArr,
    float* __restrict__ out, int n_triples)
{
    const int tid  = blockIdx.x * blockDim.x + threadIdx.x;
    const int lane = tid & 31;
    const int wave = tid >> 5;
    const int m    = lane & 15;
    const bool lo  = lane < 16;

    int  t  = wave * 16 + m;
    bool v  = t < n_triples;
    int  tc = v ? t : 0;

    int jp = idx_j_t[tc];
    int kp = idx_k_t[tc];
    int ai = idx_i_t[tc];

    float4 pj = pd[jp];
    float4 pk = pd[kp];
    float  wj = wArr[jp];
    float  wk = wArr[kp];

    float cosT = pj.x * pk.x + pj.y * pk.y + pj.z * pk.z;   // unit vectors
    float s    = 0.5f * (pj.w + pk.w);
    float pref = v ? 0.0078125f * wj * wk : 0.0f;           // 2^(1-zeta)

    float c1p = 1.0f + cosT, c1m = 1.0f - cosT;
    float c2p = c1p * c1p,   c2m = c1m * c1m;
    float c4p = c2p * c2p,   c4m = c2m * c2m;
    float angP = pref * c4p * c4p;   // lambda = +1
    float angM = pref * c4m * c4m;   // lambda = -1

    // exponent tile E[t,m] = -eta_ang*(s_t - mu_m)^2 (cols 12..15 are padding)
    v2f a;
    a.x = lo ? (-ETA_ANG) * s * s : 1.0f;
    a.y = lo ? s : 0.0f;
    float mu = (m < N_MU_ANG) ? mus_ang[m] : 0.0f;
    v2f b;
    b.x = lo ? 1.0f : (-ETA_ANG) * mu * mu;
    b.y = lo ? (2.0f * ETA_ANG) * mu : 0.0f;
    v8f c = {};
    c = __builtin_amdgcn_wmma_f32_16x16x4_f32(
            false, a, false, b, (short)0, c, false, false);

    #pragma unroll
    for (int r = 0; r < 8; ++r) {
        int   trow = r + (lo ? 0 : 8);
        // shuffles done with full EXEC (bpermute reads 0 from disabled lanes)
        float aP  = __shfl(angP, trow, 32);
        float aM  = __shfl(angM, trow, 32);
        int   aii = __shfl(ai,   trow, 32);
        if (m < N_MU_ANG) {
            float e = __expf(c[r]);
            atomicAdd(&out[aii * N_BASIS + N_RAD + m],            aP * e);
            atomicAdd(&out[aii * N_BASIS + N_RAD + N_MU_ANG + m], aM * e);
        }
    }
}

// ---------------------------------------------------------------------------
// Pass 3: normalization (x - mean) / stddev, in place on d_out
// ---------------------------------------------------------------------------
__global__ void __launch_bounds__(256) sf_norm(
    float* __restrict__ out, const float* __restrict__ mean,
    const float* __restrict__ stddev, int n)
{
    int i = blockIdx.x * blockDim.x + threadIdx.x;
    if (i < n) {
        int c = i % N_BASIS;
        out[i] = (out[i] - mean[c]) / stddev[c];
    }
}

extern "C" void kernel_launch(void* const* d_in, const int* in_sizes, int n_in,
                              void* d_out, int out_size, void* d_ws, size_t ws_size,
                              hipStream_t stream)
{
    const int*   Z       = (const int*)d_in[0];
    const float* r_ij    = (const float*)d_in[1];
    const int*   idx_i   = (const int*)d_in[2];
    const int*   idx_j   = (const int*)d_in[3];
    const int*   idx_i_t = (const int*)d_in[4];
    const int*   idx_j_t = (const int*)d_in[5];
    const int*   idx_k_t = (const int*)d_in[6];
    const float* mus_rad = (const float*)d_in[7];
    const float* mus_ang = (const float*)d_in[8];
    const float* mean    = (const float*)d_in[9];
    const float* stddev  = (const float*)d_in[10];
    float* out = (float*)d_out;

    const int n_pairs   = in_sizes[2];
    const int n_triples = in_sizes[4];

    float4* pd   = (float4*)d_ws;                                  // 16 B / pair
    float*  wArr = (float*)((char*)d_ws + (size_t)n_pairs * sizeof(float4));

    sf_zero<<<(out_size + 255) / 256, 256, 0, stream>>>(out, out_size);

    long waves_p = (n_pairs + 15) / 16;
    long thr_p   = waves_p * 32;
    sf_pairs<<<(int)((thr_p + 255) / 256), 256, 0, stream>>>(
        Z, r_ij, idx_i, idx_j, mus_rad, pd, wArr, out, n_pairs);

    long waves_t = (n_triples + 15) / 16;
    long thr_t   = waves_t * 32;
    sf_triples<<<(int)((thr_t + 255) / 256), 256, 0, stream>>>(
        idx_i_t, idx_j_t, idx_k_t, mus_ang, pd, wArr, out, n_triples);

    sf_norm<<<(out_size + 255) / 256, 256, 0, stream>>>(out, mean, stddev, out_size);
}